// MyGAT_58858231824474
// MI455X (gfx1250) — compile-verified
//
#include <hip/hip_runtime.h>
#include <hip/hip_bf16.h>

// Problem constants (from reference)
#define NN   50000     // nodes
#define NE   800000    // edges (before self loops)
#define HD   128       // hidden / input dim
#define OUTD 64        // output dim
#define NL   3         // GAT layers
#define NG   64        // graphs (pool segments)
#define EN   (NE + NN) // edges incl. self loops
#define NEG  0.2f      // leaky relu slope

typedef __attribute__((ext_vector_type(2))) float v2f;
typedef __attribute__((ext_vector_type(8))) float v8f;
typedef __attribute__((ext_vector_type(4))) int   v4i;
typedef long long i64;

#if __has_builtin(__builtin_amdgcn_global_load_async_to_lds_b128)
#define HAVE_ASYNC_LDS 1
typedef __attribute__((address_space(1))) v4i* gv4i_p;  // global src
typedef __attribute__((address_space(3))) v4i* lv4i_p;  // LDS dst
#else
#define HAVE_ASYNC_LDS 0
#endif

__device__ __forceinline__ void wait_async_lds() {
#if HAVE_ASYNC_LDS
#if __has_builtin(__builtin_amdgcn_s_wait_asynccnt)
  __builtin_amdgcn_s_wait_asynccnt(0);
#else
  asm volatile("s_wait_asynccnt 0x0" ::: "memory");
#endif
#endif
}

// ---- monotone float<->uint mapping for atomicMax-based segment max ----
__device__ __forceinline__ unsigned fkey(float f) {
  unsigned u = __float_as_uint(f);
  return (u & 0x80000000u) ? ~u : (u | 0x80000000u);
}
__device__ __forceinline__ float fdecode(unsigned k) {
  unsigned u = (k & 0x80000000u) ? (k & 0x7FFFFFFFu) : ~k;
  return __uint_as_float(u);
}

// ---------------------------------------------------------------------
// WMMA fp32 GEMM: C[M,Ncols] = A[M,K] * B[K,Ncols] (+ bias[Ncols])
// Block = 256 threads (8 waves); block computes a 16-row slab of C.
// The 16xK A tile (K<=128) is staged once into LDS via the CDNA5 async
// global->LDS copy (ASYNCcnt) and shared by all 8 waves; each wave then
// computes one 16x16 tile with V_WMMA_F32_16X16X4_F32 (K/4 steps).
// VGPR layouts per CDNA5 ISA 7.12.2:
//   A 16x4 : lane<16 -> M=lane,  K = k0 + 2*half + v   (v = vgpr 0..1)
//   B 4x16 : lane<16 -> N=lane,  K = k0 + 2*half + v
//   C 16x16: M = v + 8*half, N = lane&15               (v = vgpr 0..7)
// LDS row stride 132 floats: bank = (4*l16 + ka) % 64 -> conflict-free
// ds_load_b64 across the full wave.
// ---------------------------------------------------------------------
#define TSTR 132

__global__ void gemm16_wmma_f32(const float* __restrict__ A,
                                const float* __restrict__ B,
                                const float* __restrict__ bias,
                                float* __restrict__ C,
                                int K, int Ncols) {
  __shared__ float tileA[16 * TSTR];

  const int wave = threadIdx.x >> 5;
  const int lane = threadIdx.x & 31;
  const int half = lane >> 4;
  const int l16  = lane & 31 & 15;
  const int m0 = blockIdx.x * 16;
  const int n0 = wave * 16;
  const bool active = (n0 < Ncols);

  // ---- stage A tile (16 x K floats) into LDS, 16B chunks ----
  // chunks = 16*K/4; with K=128 -> 512 chunks; 256 threads -> 2 each
  const int nchunk = 16 * (K >> 2);
  for (int ch = threadIdx.x; ch < nchunk; ch += blockDim.x) {
    const int row = ch / (K >> 2);
    const int col = (ch % (K >> 2)) << 2;
    const float* g = A + (size_t)(m0 + row) * K + col;
#if HAVE_ASYNC_LDS
    __builtin_amdgcn_global_load_async_to_lds_b128(
        (gv4i_p)g, (lv4i_p)&tileA[row * TSTR + col], 0, 0);
#else
    float4 v = *(const float4*)g;
    *(float4*)&tileA[row * TSTR + col] = v;
#endif
  }
  wait_async_lds();
  __syncthreads();

  if (active) {
    const float* __restrict__ abase = &tileA[l16 * TSTR + 2 * half];
    const float* __restrict__ bbase = B + (size_t)(2 * half) * Ncols + n0 + l16;
    v8f acc = {};
#pragma unroll 4
    for (int k = 0; k < K; k += 4) {
      v2f av = *(const v2f*)(abase + k);
      v2f bv;
      bv.x = bbase[(size_t)k * Ncols];
      bv.y = bbase[(size_t)k * Ncols + Ncols];
      acc = __builtin_amdgcn_wmma_f32_16x16x4_f32(
          /*neg_a=*/false, av, /*neg_b=*/false, bv,
          /*c_mod=*/(short)0, acc, /*reuse_a=*/false, /*reuse_b=*/false);
    }
    const float bb = bias ? bias[n0 + l16] : 0.0f;
#pragma unroll
    for (int v = 0; v < 8; ++v)
      C[(size_t)(m0 + v + 8 * half) * Ncols + n0 + l16] = acc[v] + bb;
  }
}

// alpha_s[n] = dot(hp[n,:], a_s); alpha_d[n] = dot(hp[n,:], a_d)
__global__ void alpha_k(const float* __restrict__ hp,
                        const float* __restrict__ a_s,
                        const float* __restrict__ a_d,
                        float* __restrict__ as_out,
                        float* __restrict__ ad_out) {
  int n = blockIdx.x * blockDim.x + threadIdx.x;
  if (n >= NN) return;
  const float* row = hp + (size_t)n * HD;
  float s = 0.0f, d = 0.0f;
#pragma unroll 4
  for (int c = 0; c < HD; c += 4) {
    float4 h4 = *(const float4*)(row + c);
    float4 s4 = *(const float4*)(a_s + c);
    float4 d4 = *(const float4*)(a_d + c);
    s += h4.x * s4.x + h4.y * s4.y + h4.z * s4.z + h4.w * s4.w;
    d += h4.x * d4.x + h4.y * d4.y + h4.z * d4.z + h4.w * d4.w;
  }
  as_out[n] = s;
  ad_out[n] = d;
}

// segment max of leaky_relu(alpha_s[src]+alpha_d[dst]) over dst
__global__ void edge_max_k(const i64* __restrict__ src, const i64* __restrict__ dst,
                           const float* __restrict__ as_, const float* __restrict__ ad_,
                           unsigned* __restrict__ mkey) {
  int t = blockIdx.x * blockDim.x + threadIdx.x;
  if (t >= EN) return;
  int s, d;
  if (t < NE) { s = (int)src[t]; d = (int)dst[t]; }
  else        { s = d = t - NE; }
  float e = as_[s] + ad_[d];
  e = (e > 0.0f) ? e : NEG * e;
  atomicMax(&mkey[d], fkey(e));
}

// ex = exp(e - m[dst]); denom[dst] += ex (segment sum)
__global__ void edge_exp_k(const i64* __restrict__ src, const i64* __restrict__ dst,
                           const float* __restrict__ as_, const float* __restrict__ ad_,
                           const unsigned* __restrict__ mkey,
                           float* __restrict__ exb, float* __restrict__ denom) {
  int t = blockIdx.x * blockDim.x + threadIdx.x;
  if (t >= EN) return;
  int s, d;
  if (t < NE) { s = (int)src[t]; d = (int)dst[t]; }
  else        { s = d = t - NE; }
  float e = as_[s] + ad_[d];
  e = (e > 0.0f) ? e : NEG * e;
  float ex = __expf(e - fdecode(mkey[d]));
  exb[t] = ex;
  atomicAdd(&denom[d], ex);
}

// out[dst,:] += hp[src,:] * (ex / denom[dst]); one wave per edge,
// each lane owns 4 consecutive channels (32 lanes * 4 = 128 = HD).
__global__ void edge_agg_k(const i64* __restrict__ src, const i64* __restrict__ dst,
                           const float* __restrict__ hp,
                           const float* __restrict__ exb,
                           const float* __restrict__ denom,
                           float* __restrict__ outacc) {
  int edge = (int)((blockIdx.x * (size_t)blockDim.x + threadIdx.x) >> 5);
  if (edge >= EN) return;
  int lane = threadIdx.x & 31;
  int s, d;
  if (edge < NE) { s = (int)src[edge]; d = (int)dst[edge]; }
  else           { s = d = edge - NE; }
  float alpha = exb[edge] / denom[d];
  float4 h4 = *(const float4*)(hp + (size_t)s * HD + lane * 4);
  float* o = outacc + (size_t)d * HD + lane * 4;
  atomicAdd(o + 0, h4.x * alpha);
  atomicAdd(o + 1, h4.y * alpha);
  atomicAdd(o + 2, h4.z * alpha);
  atomicAdd(o + 3, h4.w * alpha);
}

// h = (acc + b) with optional ReLU
__global__ void bias_act_k(const float* __restrict__ acc, const float* __restrict__ b,
                           float* __restrict__ out, int do_relu) {
  size_t t = blockIdx.x * (size_t)blockDim.x + threadIdx.x;
  if (t >= (size_t)NN * HD) return;
  float v = acc[t] + b[t & (HD - 1)];
  if (do_relu) v = fmaxf(v, 0.0f);
  out[t] = v;
}

// pooled[batch[n], c] += h[n, c]
__global__ void pool_k(const float* __restrict__ h, const i64* __restrict__ batch,
                       float* __restrict__ pooled) {
  size_t t = blockIdx.x * (size_t)blockDim.x + threadIdx.x;
  if (t >= (size_t)NN * HD) return;
  int n = (int)(t >> 7);
  int c = (int)(t & (HD - 1));
  atomicAdd(&pooled[(size_t)((int)batch[n]) * HD + c], h[t]);
}

extern "C" void kernel_launch(void* const* d_in, const int* in_sizes, int n_in,
                              void* d_out, int out_size, void* d_ws, size_t ws_size,
                              hipStream_t stream) {
  const float* x     = (const float*)d_in[0];
  const i64*   ei    = (const i64*)d_in[1];
  const i64*   batch = (const i64*)d_in[2];
  const float* W0    = (const float*)d_in[3];
  const float* b0    = (const float*)d_in[4];
  const float* Wl    = (const float*)d_in[5];
  const float* a_src = (const float*)d_in[6];
  const float* a_dst = (const float*)d_in[7];
  const float* bl    = (const float*)d_in[8];
  const float* Wf    = (const float*)d_in[9];
  const float* bf    = (const float*)d_in[10];
  const i64* src = ei;
  const i64* dst = ei + NE;

  // workspace carve-up (floats, 256B aligned regions)
  auto au = [](size_t v) { return (v + 63) & ~(size_t)63; };
  float* base = (float*)d_ws;
  size_t off = 0;
  float*    hA     = base + off; off += au((size_t)NN * HD);
  float*    hp     = base + off; off += au((size_t)NN * HD);
  float*    hN     = base + off; off += au((size_t)NN * HD);
  float*    asb    = base + off; off += au((size_t)NN);
  float*    adb    = base + off; off += au((size_t)NN);
  unsigned* mkey   = (unsigned*)(base + off); off += au((size_t)NN);
  float*    denom  = base + off; off += au((size_t)NN);
  float*    exb    = base + off; off += au((size_t)EN);
  float*    pooled = base + off; off += au((size_t)NG * HD);

  const dim3 blk(256);

  // h = x @ W0 + b0
  gemm16_wmma_f32<<<NN / 16, blk, 0, stream>>>(x, W0, b0, hA, HD, HD);

  for (int l = 0; l < NL; ++l) {
    // hp = h @ Wl[l]
    gemm16_wmma_f32<<<NN / 16, blk, 0, stream>>>(hA, Wl + (size_t)l * HD * HD,
                                                 nullptr, hp, HD, HD);
    // per-node attention dots
    alpha_k<<<(NN + 255) / 256, blk, 0, stream>>>(hp, a_src + l * HD,
                                                  a_dst + l * HD, asb, adb);
    // reset segment accumulators (graph-capture-safe async memsets)
    (void)hipMemsetAsync(mkey, 0, (size_t)NN * sizeof(unsigned), stream);
    (void)hipMemsetAsync(denom, 0, (size_t)NN * sizeof(float), stream);
    (void)hipMemsetAsync(hN, 0, (size_t)NN * HD * sizeof(float), stream);

    edge_max_k<<<(EN + 255) / 256, blk, 0, stream>>>(src, dst, asb, adb, mkey);
    edge_exp_k<<<(EN + 255) / 256, blk, 0, stream>>>(src, dst, asb, adb, mkey,
                                                     exb, denom);
    edge_agg_k<<<(EN + 7) / 8, blk, 0, stream>>>(src, dst, hp, exb, denom, hN);

    // h = act(out + bl[l]) ; ReLU between layers only
    bias_act_k<<<(int)(((size_t)NN * HD + 255) / 256), blk, 0, stream>>>(
        hN, bl + l * HD, hA, (l < NL - 1) ? 1 : 0);
  }

  // global add pool, then final linear (WMMA again; idle waves still barrier)
  (void)hipMemsetAsync(pooled, 0, (size_t)NG * HD * sizeof(float), stream);
  pool_k<<<(int)(((size_t)NN * HD + 255) / 256), blk, 0, stream>>>(hA, batch,
                                                                   pooled);
  gemm16_wmma_f32<<<NG / 16, blk, 0, stream>>>(pooled, Wf, bf, (float*)d_out,
                                               HD, OUTD);
}